// ResVGAE_GCN_59974923321345
// MI455X (gfx1250) — compile-verified
//
#include <hip/hip_runtime.h>
#include <hip/hip_bf16.h>
#include <cfloat>
#include <cstddef>

// ---------------- problem constants ----------------
#define PN   163840   // nodes
#define PE   655360   // edges
#define PB   4096     // graphs / batch rows
#define PFIN 78
#define PH   128
#define PL   735
#define BN_EPS 1e-5f

typedef float v2f __attribute__((ext_vector_type(2)));
typedef float v8f __attribute__((ext_vector_type(8)));

// ===================================================================
// Degree / dinv
// ===================================================================
__global__ void k_deg(const int* __restrict__ dst, float* __restrict__ deg, int E) {
    int e = blockIdx.x * blockDim.x + threadIdx.x;
    if (e < E) atomicAdd(&deg[dst[e]], 1.0f);
}

__global__ void k_dinv(float* __restrict__ deg, int n) {
    int i = blockIdx.x * blockDim.x + threadIdx.x;
    if (i < n) deg[i] = rsqrtf(deg[i] + 1.0f);   // in-place: deg -> dinv
}

// ===================================================================
// fp32 GEMM via V_WMMA_F32_16X16X4_F32.
// C[M x Nc] = A[M x K] @ W[K x Nc] (+ bias).  M % 16 == 0, Nc % 16 == 0,
// K even (true for all layers: 78/128/256/512/1024/2944).
// One wave computes a 16x16 tile; 4 waves per block tile 64 rows.
// A frag (16x4): lanes 0-15 rows M=0..15 K={0,1}; lanes 16-31 K={2,3}.
// B frag (4x16): vgpr i -> lanes 0-15 K=i, lanes 16-31 K=i+2.
// C/D (16x16):  vgpr i -> lanes 0-15 M=i, lanes 16-31 M=i+8.
// Main loop is guard-free (K rounded down to x4); <=3 tail elements are
// handled once after the loop so the hot path has no exec-mask traffic.
// ===================================================================
__global__ void gemm_wmma_f32(const float* __restrict__ A,
                              const float* __restrict__ W,
                              const float* __restrict__ bias,
                              float* __restrict__ C,
                              int M, int K, int Nc) {
    int wave = threadIdx.x >> 5;
    int lane = threadIdx.x & 31;
    int rowTile = blockIdx.x * 4 + wave;
    int row0 = rowTile * 16;
    if (row0 >= M) return;                 // wave-uniform, EXEC stays full
    int col0 = blockIdx.y * 16;
    int lm = lane & 15;                    // 0..15
    int lh = lane >> 4;                    // 0,1

    v8f acc = {0.f, 0.f, 0.f, 0.f, 0.f, 0.f, 0.f, 0.f};
    // Arow points at this lane's k = 2*lh element of its A row (8B aligned: K even)
    const float* Arow = A + (size_t)(row0 + lm) * K + 2 * lh;
    // Wc points at row (2*lh), column (col0+lm)
    const float* Wc = W + (size_t)(2 * lh) * Nc + col0 + lm;

    int K4 = K & ~3;
    for (int k0 = 0; k0 < K4; k0 += 4) {
        v2f a = *(const v2f*)(Arow + k0);          // global_load_b64
        v2f b;
        b.x = Wc[(size_t)k0 * Nc];
        b.y = Wc[(size_t)k0 * Nc + Nc];
        acc = __builtin_amdgcn_wmma_f32_16x16x4_f32(
            false, a, false, b, (short)0, acc, false, false);
    }
    if (K4 < K) {                                  // remainder 1..3 (K=78 -> 2)
        int ka = K4 + 2 * lh;
        v2f a, b;
        a.x = (ka     < K) ? Arow[K4]                     : 0.0f;
        a.y = (ka + 1 < K) ? Arow[K4 + 1]                 : 0.0f;
        b.x = (ka     < K) ? Wc[(size_t)K4 * Nc]          : 0.0f;
        b.y = (ka + 1 < K) ? Wc[(size_t)K4 * Nc + Nc]     : 0.0f;
        acc = __builtin_amdgcn_wmma_f32_16x16x4_f32(
            false, a, false, b, (short)0, acc, false, false);
    }

    float bv = bias ? bias[col0 + lm] : 0.0f;
#pragma unroll
    for (int i = 0; i < 8; ++i) {
        C[(size_t)(row0 + i + 8 * lh) * Nc + col0 + lm] = acc[i] + bv;
    }
}

// ===================================================================
// GCN aggregation
// ===================================================================
// agg[i,c] = h[i,c]*dinv[i]^2 + bias[c]   (self-loop term + bias)
__global__ void k_init_agg(const float* __restrict__ h, const float* __restrict__ dinv,
                           const float* __restrict__ bias, float* __restrict__ agg, int n) {
    int i = blockIdx.x * blockDim.x + threadIdx.x;
    int total = n * PH;
    if (i >= total) return;
    int node = i / PH, c = i - node * PH;
    float d = dinv[node];
    agg[i] = h[i] * d * d + bias[c];
}

// one wave per edge; lane handles 4 of the 128 features
__global__ void k_scatter_edges(const float* __restrict__ h,
                                const int* __restrict__ src, const int* __restrict__ dst,
                                const float* __restrict__ dinv, float* __restrict__ agg, int E) {
    int e = blockIdx.x * 8 + (threadIdx.x >> 5);
    if (e >= E) return;
    int lane = threadIdx.x & 31;
    int s = src[e], d = dst[e];
    float w = dinv[s] * dinv[d];
    float4 v = ((const float4*)(h + (size_t)s * PH))[lane];
    float* p = agg + (size_t)d * PH + lane * 4;
    atomicAdd(p + 0, v.x * w);
    atomicAdd(p + 1, v.y * w);
    atomicAdd(p + 2, v.z * w);
    atomicAdd(p + 3, v.w * w);
}

// ===================================================================
// BatchNorm1d over rows (R rows, C channels)
// ===================================================================
__global__ void k_bn_stats(const float* __restrict__ x, float* __restrict__ sum,
                           float* __restrict__ sumsq, int R, int C, int rowsPerBlock) {
    int r0 = blockIdx.x * rowsPerBlock;
    int r1 = min(r0 + rowsPerBlock, R);
    for (int c = threadIdx.x; c < C; c += blockDim.x) {
        float s = 0.f, q = 0.f;
        for (int r = r0; r < r1; ++r) {
            float v = x[(size_t)r * C + c];
            s += v; q += v * v;
        }
        atomicAdd(&sum[c], s);
        atomicAdd(&sumsq[c], q);
    }
}

__global__ void k_bn_apply(const float* __restrict__ x, const float* __restrict__ sum,
                           const float* __restrict__ sumsq, const float* __restrict__ g,
                           const float* __restrict__ beta, const float* __restrict__ residual,
                           float* __restrict__ y, int R, int C, int doRelu) {
    size_t i = (size_t)blockIdx.x * blockDim.x + threadIdx.x;
    size_t total = (size_t)R * C;
    if (i >= total) return;
    int c = (int)(i % C);
    float invR = 1.0f / (float)R;
    float m = sum[c] * invR;
    float var = sumsq[c] * invR - m * m;
    float v = g[c] * (x[i] - m) * rsqrtf(var + BN_EPS) + beta[c];
    if (residual) v += residual[i];
    if (doRelu) v = fmaxf(v, 0.0f);
    y[i] = v;
}

// ===================================================================
// reparameterization, pooling, LayerNorm
// ===================================================================
__global__ void k_reparam(const float* __restrict__ mu, const float* __restrict__ lv,
                          const float* __restrict__ eps, float* __restrict__ z, int total) {
    int i = blockIdx.x * blockDim.x + threadIdx.x;
    if (i < total) z[i] = mu[i] + eps[i] * __expf(0.5f * lv[i]);
}

__global__ void k_zp_scatter(const float* __restrict__ z, const int* __restrict__ batch,
                             float* __restrict__ zp, int n) {
    int node = blockIdx.x * 8 + (threadIdx.x >> 5);
    if (node >= n) return;
    int lane = threadIdx.x & 31;
    int b = batch[node];
    float4 v = ((const float4*)(z + (size_t)node * PH))[lane];
    float* p = zp + (size_t)b * PH + lane * 4;
    atomicAdd(p + 0, v.x);
    atomicAdd(p + 1, v.y);
    atomicAdd(p + 2, v.z);
    atomicAdd(p + 3, v.w);
}

// LayerNorm over H=128 per row; one wave per row
__global__ void k_ln(const float* __restrict__ zp, const float* __restrict__ g,
                     const float* __restrict__ beta, float* __restrict__ out, int R) {
    int row = blockIdx.x * 8 + (threadIdx.x >> 5);
    if (row >= R) return;
    int lane = threadIdx.x & 31;
    float4 v = ((const float4*)(zp + (size_t)row * PH))[lane];
    float s = v.x + v.y + v.z + v.w;
    float q = v.x * v.x + v.y * v.y + v.z * v.z + v.w * v.w;
    for (int off = 16; off; off >>= 1) {
        s += __shfl_xor(s, off);
        q += __shfl_xor(q, off);
    }
    float m = s * (1.0f / PH);
    float var = q * (1.0f / PH) - m * m;
    float inv = rsqrtf(var + BN_EPS);
    int c0 = lane * 4;
    float4 r;
    r.x = g[c0 + 0] * (v.x - m) * inv + beta[c0 + 0];
    r.y = g[c0 + 1] * (v.y - m) * inv + beta[c0 + 1];
    r.z = g[c0 + 2] * (v.z - m) * inv + beta[c0 + 2];
    r.w = g[c0 + 3] * (v.w - m) * inv + beta[c0 + 3];
    ((float4*)(out + (size_t)row * PH))[lane] = r;
}

// ===================================================================
// Conv1d (K=8, VALID) branch.
// Pass 1: per-channel sum/sumsq of conv output over (B, OL)  (recompute)
// Pass 2: recompute conv, apply BN+ReLU, maxpool3 -> pooled output
// ===================================================================
__global__ void k_conv_stats(const float* __restrict__ in, const float* __restrict__ Wk,
                             const float* __restrict__ bias, float* __restrict__ sum,
                             float* __restrict__ sumsq, int IC, int OC, int INL, int OL) {
    __shared__ float s_sum[128];
    __shared__ float s_sq[128];
    int b = blockIdx.x;
    for (int i = threadIdx.x; i < OC; i += blockDim.x) { s_sum[i] = 0.f; s_sq[i] = 0.f; }
    __syncthreads();
    const float* inb = in + (size_t)b * IC * INL;
    for (int idx = threadIdx.x; idx < OC * OL; idx += blockDim.x) {
        int oc = idx / OL, ol = idx - oc * OL;
        float y = bias[oc];
        const float* w = Wk + (size_t)oc * IC * 8;
        for (int ic = 0; ic < IC; ++ic) {
            const float* ip = inb + (size_t)ic * INL + ol;
            const float* wp = w + ic * 8;
#pragma unroll
            for (int k = 0; k < 8; ++k) y = fmaf(wp[k], ip[k], y);
        }
        atomicAdd(&s_sum[oc], y);
        atomicAdd(&s_sq[oc], y * y);
    }
    __syncthreads();
    for (int i = threadIdx.x; i < OC; i += blockDim.x) {
        atomicAdd(&sum[i], s_sum[i]);
        atomicAdd(&sumsq[i], s_sq[i]);
    }
}

__global__ void k_conv_bn_relu_pool(const float* __restrict__ in, const float* __restrict__ Wk,
                                    const float* __restrict__ bias, const float* __restrict__ sum,
                                    const float* __restrict__ sumsq, const float* __restrict__ g,
                                    const float* __restrict__ beta, float* __restrict__ out,
                                    int IC, int OC, int INL, int OP, float invCount) {
    int b = blockIdx.x;
    const float* inb = in + (size_t)b * IC * INL;
    for (int idx = threadIdx.x; idx < OC * OP; idx += blockDim.x) {
        int oc = idx / OP, op = idx - oc * OP;
        float m = sum[oc] * invCount;
        float var = sumsq[oc] * invCount - m * m;
        float sc = g[oc] * rsqrtf(var + BN_EPS);
        float bb = beta[oc];
        const float* w = Wk + (size_t)oc * IC * 8;
        float best = -FLT_MAX;
#pragma unroll
        for (int j = 0; j < 3; ++j) {
            int ol = 3 * op + j;
            float y = bias[oc];
            for (int ic = 0; ic < IC; ++ic) {
                const float* ip = inb + (size_t)ic * INL + ol;
                const float* wp = w + ic * 8;
#pragma unroll
                for (int k = 0; k < 8; ++k) y = fmaf(wp[k], ip[k], y);
            }
            y = fmaxf(sc * (y - m) + bb, 0.0f);
            best = fmaxf(best, y);
        }
        out[(size_t)b * OC * OP + (size_t)oc * OP + op] = best;
    }
}

// ===================================================================
// misc small kernels
// ===================================================================
__global__ void k_concat(const float* __restrict__ zp, const float* __restrict__ xt,
                         float* __restrict__ xc, int R) {
    int i = blockIdx.x * blockDim.x + threadIdx.x;
    int total = R * 256;
    if (i >= total) return;
    int r = i >> 8, c = i & 255;
    xc[i] = (c < 128) ? zp[(size_t)r * 128 + c] : xt[(size_t)r * 128 + (c - 128)];
}

// final 128 -> 1 linear; one wave per row
__global__ void k_fc_out(const float* __restrict__ x, const float* __restrict__ w,
                         const float* __restrict__ bias, float* __restrict__ out, int R) {
    int row = blockIdx.x * 8 + (threadIdx.x >> 5);
    if (row >= R) return;
    int lane = threadIdx.x & 31;
    const float* xr = x + (size_t)row * 128;
    float s = 0.f;
#pragma unroll
    for (int j = 0; j < 4; ++j) { int c = lane + 32 * j; s = fmaf(xr[c], w[c], s); }
    for (int off = 16; off; off >>= 1) s += __shfl_xor(s, off);
    if (lane == 0) out[row] = s + bias[0];
}

// ===================================================================
// host side
// ===================================================================
extern "C" void kernel_launch(void* const* d_in, const int* in_sizes, int n_in,
                              void* d_out, int out_size, void* d_ws, size_t ws_size,
                              hipStream_t stream) {
    (void)in_sizes; (void)n_in; (void)out_size; (void)ws_size;

    // ---- inputs (setup_inputs dict order, params flattened in file order) ----
    const float* x      = (const float*)d_in[0];   // N x 78
    const float* target = (const float*)d_in[1];   // B x 735
    const float* epsin  = (const float*)d_in[2];   // N x 128
    int p = 3;
    const float* gcnW[4]; const float* gcnB[4]; const float* bnG[4]; const float* bnB[4];
    for (int l = 0; l < 4; ++l) {
        gcnW[l] = (const float*)d_in[p++]; gcnB[l] = (const float*)d_in[p++];
        bnG[l]  = (const float*)d_in[p++]; bnB[l]  = (const float*)d_in[p++];
    }
    const float* resW = (const float*)d_in[p++]; const float* resB = (const float*)d_in[p++];
    const float* muW  = (const float*)d_in[p++]; const float* muB  = (const float*)d_in[p++];
    const float* lvW  = (const float*)d_in[p++]; const float* lvB  = (const float*)d_in[p++];
    const float* lnG  = (const float*)d_in[p++]; const float* lnB  = (const float*)d_in[p++];
    const float* cK1 = (const float*)d_in[p++]; const float* cB1 = (const float*)d_in[p++];
    const float* xG1 = (const float*)d_in[p++]; const float* xB1 = (const float*)d_in[p++];
    const float* cK2 = (const float*)d_in[p++]; const float* cB2 = (const float*)d_in[p++];
    const float* xG2 = (const float*)d_in[p++]; const float* xB2 = (const float*)d_in[p++];
    const float* cK3 = (const float*)d_in[p++]; const float* cB3 = (const float*)d_in[p++];
    const float* xG3 = (const float*)d_in[p++]; const float* xB3 = (const float*)d_in[p++];
    const float* fxtW = (const float*)d_in[p++]; const float* fxtB = (const float*)d_in[p++];
    const float* f1W = (const float*)d_in[p++]; const float* f1B = (const float*)d_in[p++];
    const float* bf1G = (const float*)d_in[p++]; const float* bf1B = (const float*)d_in[p++];
    const float* f2W = (const float*)d_in[p++]; const float* f2B = (const float*)d_in[p++];
    const float* bf2G = (const float*)d_in[p++]; const float* bf2B = (const float*)d_in[p++];
    const float* f3W = (const float*)d_in[p++]; const float* f3B = (const float*)d_in[p++];
    const float* bf3G = (const float*)d_in[p++]; const float* bf3B = (const float*)d_in[p++];
    const float* outW = (const float*)d_in[p++]; const float* outBias = (const float*)d_in[p++];
    const int* edges = (const int*)d_in[p++];     // 2 x E
    const int* batch = (const int*)d_in[p++];     // N
    const int* srcI = edges;
    const int* dstI = edges + PE;

    // ---- workspace layout (floats) ----
    float* W = (float*)d_ws;
    size_t o = 0;
    float* stSum = W + o; o += 1024;
    float* stSq  = W + o; o += 1024;
    float* dinv  = W + o; o += PN;                     // deg -> dinv in place
    float* zp    = W + o; o += (size_t)PB * PH;
    float* xt    = W + o; o += (size_t)PB * PH;
    float* xc    = W + o; o += (size_t)PB * 256;
    float* fb1   = W + o; o += (size_t)PB * 1024;
    float* fb2   = W + o; o += (size_t)PB * 512;
    float* fb3   = W + o; o += (size_t)PB * PH;
    const size_t NH = (size_t)PN * PH;
    float* bigA = W + o;            // identity, later z; conv scratch reuses bigA..
    float* bigB = bigA + NH;
    float* bigC = bigB + NH;
    float* bigD = bigC + NH;
    // conv-branch reuse of the 4*NH region (after z has been pooled into zp)
    float* p1 = bigA;                               // B*32*242 = 31,719,424
    float* p2 = p1 + (size_t)PB * 32 * 242;         // B*64*78  = 20,447,232
    float* t3 = p2 + (size_t)PB * 64 * 78;          // B*128*23 = 12,058,624

    float* outHead = (float*)d_out;        // B x 1
    float* zpOut   = outHead + PB;         // B x 128 (post-LN zp, also fed to concat)

    auto gemm = [&](const float* A, const float* Wm, const float* bias, float* C,
                    int M, int K, int Nc) {
        dim3 g((M + 63) / 64, Nc / 16);
        gemm_wmma_f32<<<g, dim3(128), 0, stream>>>(A, Wm, bias, C, M, K, Nc);
    };
    auto bnZero = [&]() { (void)hipMemsetAsync(stSum, 0, 2 * 1024 * sizeof(float), stream); };
    auto bn = [&](const float* xin, const float* g_, const float* b_, const float* res,
                  float* y, int R, int C, int rpb) {
        bnZero();
        int bd = (C >= 256) ? 256 : 128;
        k_bn_stats<<<dim3((R + rpb - 1) / rpb), dim3(bd), 0, stream>>>(xin, stSum, stSq, R, C, rpb);
        size_t tot = (size_t)R * C;
        k_bn_apply<<<dim3((unsigned)((tot + 255) / 256)), dim3(256), 0, stream>>>(
            xin, stSum, stSq, g_, b_, res, y, R, C, 1);
    };

    // ---- 1. degree / dinv ----
    (void)hipMemsetAsync(dinv, 0, PN * sizeof(float), stream);
    k_deg<<<dim3((PE + 255) / 256), dim3(256), 0, stream>>>(dstI, dinv, PE);
    k_dinv<<<dim3((PN + 255) / 256), dim3(256), 0, stream>>>(dinv, PN);

    // ---- 2. identity = x @ resW + resB ----
    gemm(x, resW, resB, bigA, PN, PFIN, PH);

    // ---- 3. four GCN layers ----
    const float* cur = x;
    int curK = PFIN;
    for (int l = 0; l < 4; ++l) {
        gemm(cur, gcnW[l], nullptr, bigB, PN, curK, PH);                 // h = cur @ W
        k_init_agg<<<dim3((unsigned)((NH + 255) / 256)), dim3(256), 0, stream>>>(
            bigB, dinv, gcnB[l], bigC, PN);                              // self term + bias
        k_scatter_edges<<<dim3(PE / 8), dim3(256), 0, stream>>>(bigB, srcI, dstI, dinv, bigC, PE);
        bn(bigC, bnG[l], bnB[l], (l == 3) ? bigA : nullptr, bigD, PN, PH, 256);
        cur = bigD; curK = PH;
    }

    // ---- 4. mu / logvar / z ----
    gemm(bigD, muW, muB, bigB, PN, PH, PH);
    gemm(bigD, lvW, lvB, bigC, PN, PH, PH);
    k_reparam<<<dim3((unsigned)((NH + 255) / 256)), dim3(256), 0, stream>>>(
        bigB, bigC, epsin, bigA, (int)NH);                               // z -> bigA

    // ---- 5. graph pooling + LayerNorm (zp written into d_out slice) ----
    (void)hipMemsetAsync(zp, 0, (size_t)PB * PH * sizeof(float), stream);
    k_zp_scatter<<<dim3(PN / 8), dim3(256), 0, stream>>>(bigA, batch, zp, PN);
    k_ln<<<dim3(PB / 8), dim3(256), 0, stream>>>(zp, lnG, lnB, zpOut, PB);

    // ---- 6. conv context branch (big region is now free) ----
    // conv1: 1 -> 32, INL 735, OL 728, pool -> 242
    bnZero();
    k_conv_stats<<<dim3(PB), dim3(256), 0, stream>>>(target, cK1, cB1, stSum, stSq, 1, 32, 735, 728);
    k_conv_bn_relu_pool<<<dim3(PB), dim3(256), 0, stream>>>(
        target, cK1, cB1, stSum, stSq, xG1, xB1, p1, 1, 32, 735, 242, 1.0f / ((float)PB * 728.0f));
    // conv2: 32 -> 64, INL 242, OL 235, pool -> 78
    bnZero();
    k_conv_stats<<<dim3(PB), dim3(256), 0, stream>>>(p1, cK2, cB2, stSum, stSq, 32, 64, 242, 235);
    k_conv_bn_relu_pool<<<dim3(PB), dim3(256), 0, stream>>>(
        p1, cK2, cB2, stSum, stSq, xG2, xB2, p2, 32, 64, 242, 78, 1.0f / ((float)PB * 235.0f));
    // conv3: 64 -> 128, INL 78, OL 71, pool -> 23
    bnZero();
    k_conv_stats<<<dim3(PB), dim3(256), 0, stream>>>(p2, cK3, cB3, stSum, stSq, 64, 128, 78, 71);
    k_conv_bn_relu_pool<<<dim3(PB), dim3(256), 0, stream>>>(
        p2, cK3, cB3, stSum, stSq, xG3, xB3, t3, 64, 128, 78, 23, 1.0f / ((float)PB * 71.0f));

    // ---- 7. fc1_xt: (B x 2944) @ (2944 x 128) ----
    gemm(t3, fxtW, fxtB, xt, PB, 2944, PH);

    // ---- 8. MLP head ----
    k_concat<<<dim3(PB), dim3(256), 0, stream>>>(zpOut, xt, xc, PB);     // B x 256
    gemm(xc, f1W, f1B, fb1, PB, 256, 1024);
    bn(fb1, bf1G, bf1B, nullptr, fb1, PB, 1024, 64);
    gemm(fb1, f2W, f2B, fb2, PB, 1024, 512);
    bn(fb2, bf2G, bf2B, nullptr, fb2, PB, 512, 64);
    gemm(fb2, f3W, f3B, fb3, PB, 512, 128);
    bn(fb3, bf3G, bf3B, nullptr, fb3, PB, 128, 64);
    k_fc_out<<<dim3(PB / 8), dim3(256), 0, stream>>>(fb3, outW, outBias, outHead, PB);
}